// LightGCN_6657199309528
// MI455X (gfx1250) — compile-verified
//
#include <hip/hip_runtime.h>
#include <hip/hip_bf16.h>

typedef __attribute__((ext_vector_type(2))) float v2f;
typedef __attribute__((ext_vector_type(8))) float v8f;

#define EMB_D 64

// d_out = 0.25 * concat(user,item); cur = concat(user,item)
__global__ void lgcn_init(const float* __restrict__ ue, const float* __restrict__ ie,
                          float* __restrict__ cur, float* __restrict__ acc,
                          size_t userElems, size_t total) {
    size_t i = ((size_t)blockIdx.x * blockDim.x + threadIdx.x) * 4;
    if (i >= total) return;
    float4 e = (i < userElems) ? *(const float4*)(ue + i)
                               : *(const float4*)(ie + (i - userElems));
    *(float4*)(cur + i) = e;
    float4 s;
    s.x = e.x * 0.25f; s.y = e.y * 0.25f; s.z = e.z * 0.25f; s.w = e.w * 0.25f;
    *(float4*)(acc + i) = s;
}

__global__ void lgcn_zero(float* __restrict__ p, size_t total) {
    size_t i = ((size_t)blockIdx.x * blockDim.x + threadIdx.x) * 4;
    if (i >= total) return;
    float4 z; z.x = z.y = z.z = z.w = 0.0f;
    *(float4*)(p + i) = z;
}

// COO SpMM scatter: 16 threads per edge, float4 per thread (full 256B row gather),
// native f32 atomics into L2 (y fits in 192MB L2).
__global__ void lgcn_spmm(const int* __restrict__ rows, const int* __restrict__ cols,
                          const float* __restrict__ vals, const float* __restrict__ x,
                          float* __restrict__ y, int nEdges) {
    long long t = (long long)blockIdx.x * blockDim.x + threadIdx.x;
    int e   = (int)(t >> 4);
    int sub = (int)(t & 15);
    if (e >= nEdges) return;
    int   r = rows[e];
    int   c = cols[e];
    float v = vals[e];
    float4 g = *((const float4*)(x + (size_t)c * EMB_D) + sub);
    float* yp = y + (size_t)r * EMB_D + (size_t)sub * 4;
    unsafeAtomicAdd(yp + 0, g.x * v);
    unsafeAtomicAdd(yp + 1, g.y * v);
    unsafeAtomicAdd(yp + 2, g.z * v);
    unsafeAtomicAdd(yp + 3, g.w * v);
}

// acc(16x16 tile) += 0.25 * nxt(16x16 tile), computed on the matrix pipe:
// D = A(16x4 chunk of nxt) x B(4x16 slice of 0.25*I16) + C, chained over 4 chunks.
// Layouts per CDNA5 ISA 7.12.2:
//   A: VGPR v, lane -> A[lane%16][K = v + 2*(lane>=16)]
//   B: VGPR v, lane -> B[K = v + 2*(lane>=16)][lane%16]
//   C/D: VGPR v, lane -> C[v + 8*(lane>=16)][lane%16]
__global__ void lgcn_wmma_axpy(const float* __restrict__ nxt, float* __restrict__ acc,
                               int nTiles) {
    int wave = (int)(((size_t)blockIdx.x * blockDim.x + threadIdx.x) >> 5);
    if (wave >= nTiles) return;              // wave-uniform: EXEC stays all-ones inside
    int lane = threadIdx.x & 31;
    int lo = lane & 15;
    int hi = lane >> 4;
    int tM = wave >> 2;                       // 16-row tile index
    int n0 = (wave & 3) << 4;                 // column tile: 0,16,32,48
    size_t row0 = (size_t)tM << 4;

    const float* aBase = nxt + (row0 + (size_t)lo) * EMB_D + n0;           // A row
    float*       cBase = acc + (row0 + (size_t)(8 * hi)) * EMB_D + n0 + lo; // C column slot

    v8f c;
#pragma unroll
    for (int v = 0; v < 8; ++v) c[v] = cBase[(size_t)v * EMB_D];

#pragma unroll
    for (int k = 0; k < 4; ++k) {
        int col = 4 * k + 2 * hi;             // local tile column for this lane's K pair
        v2f a, b;
        a[0] = aBase[col + 0];
        a[1] = aBase[col + 1];
        b[0] = (lo == col + 0) ? 0.25f : 0.0f;  // B = rows 4k..4k+3 of 0.25*I16
        b[1] = (lo == col + 1) ? 0.25f : 0.0f;
        c = __builtin_amdgcn_wmma_f32_16x16x4_f32(
                /*neg_a=*/false, a, /*neg_b=*/false, b,
                /*c_mod=*/(short)0, c, /*reuse_a=*/false, /*reuse_b=*/false);
    }

#pragma unroll
    for (int v = 0; v < 8; ++v) cBase[(size_t)v * EMB_D] = c[v];
}

extern "C" void kernel_launch(void* const* d_in, const int* in_sizes, int n_in,
                              void* d_out, int out_size, void* d_ws, size_t ws_size,
                              hipStream_t stream) {
    const float* ue   = (const float*)d_in[0];
    const float* ie   = (const float*)d_in[1];
    const int*   rows = (const int*)d_in[2];
    const int*   cols = (const int*)d_in[3];
    const float* vals = (const float*)d_in[4];
    float*       out  = (float*)d_out;

    size_t userElems = (size_t)in_sizes[0];
    size_t itemElems = (size_t)in_sizes[1];
    int    nEdges    = in_sizes[2];
    size_t total     = userElems + itemElems;   // N * 64 = 19,200,000 floats

    float* buf0 = (float*)d_ws;                 // ping
    float* buf1 = buf0 + total;                 // pong (needs 2*76.8MB of ws)

    int vecBlocks = (int)((total / 4 + 255) / 256);
    lgcn_init<<<vecBlocks, 256, 0, stream>>>(ue, ie, buf0, out, userElems, total);

    int nTiles     = (int)(total / 256);        // 16x16 f32 tiles = 75,000
    int wmmaBlocks = (nTiles + 7) / 8;          // 8 waves per 256-thread block
    long long spmmThreads = (long long)nEdges * 16;
    int spmmBlocks = (int)((spmmThreads + 255) / 256);

    float* cur = buf0;
    float* nxt = buf1;
    for (int layer = 0; layer < 3; ++layer) {
        lgcn_zero<<<vecBlocks, 256, 0, stream>>>(nxt, total);
        lgcn_spmm<<<spmmBlocks, 256, 0, stream>>>(rows, cols, vals, cur, nxt, nEdges);
        lgcn_wmma_axpy<<<wmmaBlocks, 256, 0, stream>>>(nxt, out, nTiles);
        float* tmp = cur; cur = nxt; nxt = tmp;
    }
}